// GraphConv_84499186582211
// MI455X (gfx1250) — compile-verified
//
#include <hip/hip_runtime.h>
#include <hip/hip_bf16.h>

#define N_NODES 100000
#define N_EDGES 600000
#define D_IN    128
#define D_OUT   128

typedef __attribute__((ext_vector_type(2))) float v2f;
typedef __attribute__((ext_vector_type(8))) float v8f;

// ---------------------------------------------------------------------------
// Kernel 1: zero agg[N_NODES*128] and deg[N_NODES] (contiguous in d_ws)
// ---------------------------------------------------------------------------
__global__ void gc_zero_kernel(float4* __restrict__ p, size_t n4) {
    size_t i = (size_t)blockIdx.x * blockDim.x + threadIdx.x;
    size_t stride = (size_t)gridDim.x * blockDim.x;
    float4 z; z.x = 0.f; z.y = 0.f; z.z = 0.f; z.w = 0.f;
    for (; i < n4; i += stride) p[i] = z;
}

// ---------------------------------------------------------------------------
// Kernel 2: edge scatter. 32 threads per edge; each thread moves a float4
// chunk of the 128-wide feature row: agg[dst] += features[src].
// features (51MB) and agg (51MB) both fit in the 192MB L2, so the gather
// and the f32 atomics are L2-resident after first touch.
// ---------------------------------------------------------------------------
__global__ void gc_scatter_kernel(const float* __restrict__ feat,
                                  const int* __restrict__ src,
                                  const int* __restrict__ dst,
                                  float* __restrict__ agg,
                                  float* __restrict__ deg) {
    long long g = (long long)blockIdx.x * blockDim.x + threadIdx.x;
    int e = (int)(g >> 5);        // edge index
    int c = (int)(g & 31);        // float4 chunk 0..31
    if (e >= N_EDGES) return;
    int s = src[e];
    int d = dst[e];
    const float4 v = ((const float4*)(feat + (size_t)s * D_IN))[c];
    float* out = agg + (size_t)d * D_IN + c * 4;
    atomicAdd(out + 0, v.x);
    atomicAdd(out + 1, v.y);
    atomicAdd(out + 2, v.z);
    atomicAdd(out + 3, v.w);
    if (c == 0) atomicAdd(deg + d, 1.0f);
}

// ---------------------------------------------------------------------------
// Kernel 3: h = (agg + feat) * rsqrt(max(deg,1));  out = h @ W^T + b
// Block: 256 threads (8 wave32), tile 64 rows x 128 cols.
// LDS: hs[64][132] (pad -> conflict-free v2f A reads),
//      Ws[128][34]  (K-slab of W, even pad -> 8B aligned v2f B reads).
// Each wave: one 16-row M-tile x four 16-col N-tiles, fp32 WMMA 16x16x4.
// A layout: lane<16 holds K={0,1}, lane>=16 holds K={2,3}, M = lane&15.
// B layout (mirror): N = lane&15, same K split.
// C/D layout: VGPR v -> M = v + 8*(lane>=16), N = lane&15.
// ---------------------------------------------------------------------------
__global__ void gc_gemm_kernel(const float* __restrict__ agg,
                               const float* __restrict__ feat,
                               const float* __restrict__ deg,
                               const float* __restrict__ W,
                               const float* __restrict__ bias,
                               float* __restrict__ out) {
    __shared__ float hs[64][132];
    __shared__ float Ws[128][34];

    const int tid = threadIdx.x;
    const int m0  = blockIdx.x * 64;

    // ---- stage h tile: 64 x 128, coalesced 512B rows ----
    {
        int col = tid & 127;
        for (int r = (tid >> 7); r < 64; r += 2) {
            int row = m0 + r;
            float v = 0.0f;
            if (row < N_NODES) {
                float nrm = rsqrtf(fmaxf(deg[row], 1.0f));
                v = (agg[(size_t)row * D_IN + col] +
                     feat[(size_t)row * D_IN + col]) * nrm;
            }
            hs[r][col] = v;
        }
    }

    const int lane = tid & 31;
    const int wave = tid >> 5;
    const int mt   = wave >> 1;              // M-tile 0..3
    const int nh   = wave & 1;               // N half 0..1
    const int l15  = lane & 15;
    const int kb2  = (lane >> 4) << 1;       // K sub-offset 0 or 2
    const int rowA = (mt << 4) + l15;        // LDS row for A

    v8f acc[4] = {};

    for (int kb = 0; kb < D_IN; kb += 32) {
        __syncthreads();
        // ---- stage W K-slab: Ws[n][kl] = W[n][kb+kl], coalesced 128B rows
        {
            int kl = tid & 31;
            for (int n = (tid >> 5); n < D_OUT; n += 8)
                Ws[n][kl] = W[(size_t)n * D_IN + kb + kl];
        }
        __syncthreads();

        #pragma unroll
        for (int kk = 0; kk < 32; kk += 4) {
            const int kloc = kk + kb2;
            v2f a = *(const v2f*)&hs[rowA][kloc];
            #pragma unroll
            for (int j = 0; j < 4; ++j) {
                const int n = (((nh << 2) + j) << 4) + l15;
                v2f bb = *(const v2f*)&Ws[n][kloc];
                acc[j] = __builtin_amdgcn_wmma_f32_16x16x4_f32(
                    /*neg_a=*/false, a, /*neg_b=*/false, bb,
                    /*c_mod=*/(short)0, acc[j],
                    /*reuse_a=*/false, /*reuse_b=*/false);
            }
        }
    }

    // ---- write D + bias ----
    const int mrow_base = m0 + (mt << 4) + ((lane >> 4) << 3);
    #pragma unroll
    for (int j = 0; j < 4; ++j) {
        const int n0 = (((nh << 2) + j) << 4);
        const float bv = bias[n0 + l15];
        #pragma unroll
        for (int v = 0; v < 8; ++v) {
            const int row = mrow_base + v;
            if (row < N_NODES)
                out[(size_t)row * D_OUT + n0 + l15] = acc[j][v] + bv;
        }
    }
}

// ---------------------------------------------------------------------------
extern "C" void kernel_launch(void* const* d_in, const int* in_sizes, int n_in,
                              void* d_out, int out_size, void* d_ws, size_t ws_size,
                              hipStream_t stream) {
    const float* feat = (const float*)d_in[0];
    const int*   src  = (const int*)d_in[1];
    const int*   dst  = (const int*)d_in[2];
    const float* W    = (const float*)d_in[3];
    const float* bias = (const float*)d_in[4];
    float* out = (float*)d_out;

    float* agg = (float*)d_ws;                       // N_NODES * 128 floats
    float* deg = agg + (size_t)N_NODES * D_IN;       // N_NODES floats

    // zero agg + deg (N_NODES*129 floats, divisible by 4)
    size_t n4 = ((size_t)N_NODES * (D_IN + 1)) / 4;
    gc_zero_kernel<<<2048, 256, 0, stream>>>((float4*)d_ws, n4);

    // scatter: 32 threads per edge
    long long nthreads = (long long)N_EDGES * 32;
    int sblocks = (int)((nthreads + 255) / 256);
    gc_scatter_kernel<<<sblocks, 256, 0, stream>>>(feat, src, dst, agg, deg);

    // fused norm + GEMM
    int gblocks = (N_NODES + 63) / 64;
    gc_gemm_kernel<<<gblocks, 256, 0, stream>>>(agg, feat, deg, W, bias, out);
}